// HyperbolicMultiheadAttention_5523327943259
// MI455X (gfx1250) — compile-verified
//
#include <hip/hip_runtime.h>

typedef __attribute__((ext_vector_type(2))) float v2f;
typedef __attribute__((ext_vector_type(8))) float v8f;
typedef __attribute__((ext_vector_type(4))) int v4i;

typedef __attribute__((address_space(1))) v4i* gas_ptr;  // global, 16B granule
typedef __attribute__((address_space(3))) v4i* las_ptr;  // LDS, 16B granule

#define E 768
#define S 197
#define B 4
#define H 12
#define HD 64
#define MROWS (B * S)   // 788
#define NKP 208         // S padded to 13*16
#define NT 13
#define KC 32
#define NSTAGE (E / KC) // 24
#define APAD 36         // row stride: 144B (16B aligned), conflict-free reads
#define KPAD 68         // row stride: 272B (16B aligned), conflict-free reads
#define NEG_BIG (-1.0e30f)

// --------------------------------------------------------------------------
// fp32 WMMA: D(16x16) = A(16x4) * B(4x16) + C   (V_WMMA_F32_16X16X4_F32)
// A frag: lane l holds row m=l&15, k = (l>=16?2:0)+reg
// B frag: lane l holds col n=l&15, k = (l>=16?2:0)+reg
// C frag: lane l holds col n=l&15, rows m = (l>=16?8:0)+reg (8 regs)
// --------------------------------------------------------------------------
static __device__ __forceinline__ v8f wmma_f32_4(v2f a, v2f b, v8f c) {
  return __builtin_amdgcn_wmma_f32_16x16x4_f32(false, a, false, b, (short)0, c,
                                               false, false);
}

// Async global -> LDS copy of 16 bytes (GLOBAL_LOAD_ASYNC_TO_LDS_B128).
// Generic->AS casts go through integers; HW truncates LDS addrs to [31:0].
static __device__ __forceinline__ void async_b128(const void* g, void* l) {
  __builtin_amdgcn_global_load_async_to_lds_b128(
      (gas_ptr)(uintptr_t)g, (las_ptr)(uint32_t)(uintptr_t)l, 0, 0);
}

static __device__ __forceinline__ void wait_async_4() {
  asm volatile("s_wait_asynccnt 0x4" ::: "memory");
}
static __device__ __forceinline__ void wait_async_0() {
  asm volatile("s_wait_asynccnt 0x0" ::: "memory");
}
// Intra-wave LDS ordering fence (DS ops are in-order per wave).
static __device__ __forceinline__ void lds_fence() {
  asm volatile("s_wait_dscnt 0x0" ::: "memory");
}

// --------------------------------------------------------------------------
// C[M,768] = A[M,768] @ W[768,768]^T   (row-major A and W; W is out x in)
// 128 threads = 4 waves; block tile 32(M) x 32(N); wave tile 16x16.
// Double-buffered async-to-LDS staging; exactly 4 async instrs/wave/stage.
// --------------------------------------------------------------------------
__global__ __launch_bounds__(128) void gemm_xwT_kernel(
    const float* __restrict__ A, const float* __restrict__ W,
    float* __restrict__ C) {
  __shared__ float As[2][32][APAD];
  __shared__ float Ws[2][32][APAD];
  const int tid = threadIdx.x;
  const int lane = tid & 31;
  const int wave = tid >> 5;
  const int m0 = blockIdx.y * 32;
  const int n0 = blockIdx.x * 32;
  const int wm = (wave >> 1) * 16;
  const int wn = (wave & 1) * 16;
  const int mA = wm + (lane & 15);
  const int nB = wn + (lane & 15);
  const int koff = (lane >> 4) * 2;

  // Stage one 32x32 k-slice into buffer `buf`. Rows past MROWS are clamped
  // (not skipped) so every wave issues exactly 4 async ops per stage; the
  // resulting garbage C rows are discarded by the guarded store.
  auto stage = [&](int t, int buf) {
    const int k0 = t * KC;
    for (int i = tid * 4; i < 32 * KC; i += 128 * 4) {  // 2 iterations
      const int r = i >> 5, c = i & 31;
      int gm = m0 + r;
      gm = (gm < MROWS) ? gm : (MROWS - 1);
      async_b128(&A[gm * E + k0 + c], &As[buf][r][c]);
      async_b128(&W[(n0 + r) * E + k0 + c], &Ws[buf][r][c]);
    }
  };

  stage(0, 0);
  v8f acc = {};
  for (int t = 0; t < NSTAGE; ++t) {
    if (t + 1 < NSTAGE) {
      stage(t + 1, (t + 1) & 1);
      wait_async_4();  // oldest stage drained, newest still in flight
    } else {
      wait_async_0();
    }
    __syncthreads();
    const float(*Ab)[APAD] = As[t & 1];
    const float(*Wb)[APAD] = Ws[t & 1];
#pragma unroll
    for (int kk = 0; kk < KC; kk += 4) {
      v2f a, b;
      a.x = Ab[mA][kk + koff];
      a.y = Ab[mA][kk + koff + 1];
      b.x = Wb[nB][kk + koff];
      b.y = Wb[nB][kk + koff + 1];
      acc = wmma_f32_4(a, b, acc);
    }
    __syncthreads();
  }
  const int n = n0 + wn + (lane & 15);
  const int rb = (lane >> 4) * 8;
#pragma unroll
  for (int r = 0; r < 8; ++r) {
    const int m = m0 + wm + rb + r;
    if (m < MROWS) C[m * E + n] = acc[r];
  }
}

// --------------------------------------------------------------------------
// Per-row mobius_matvec normalization + mobius_add(bias).
// One 256-thread block per row (768 = 3 * 256 elements/thread).
// Safe for MX == Out (in-place).
// --------------------------------------------------------------------------
__global__ __launch_bounds__(256) void mobius_kernel(
    const float* __restrict__ Xin, const float* MX,
    const float* __restrict__ bias, float* Out) {
  const int row = blockIdx.x;
  const int tid = threadIdx.x;
  const int lane = tid & 31;
  const int wave = tid >> 5;
  __shared__ float red[4][8];
  __shared__ float sc[4];
  const float* xr = Xin + (size_t)row * E;
  const float* mr = MX + (size_t)row * E;
  const int i0 = tid, i1 = tid + 256, i2 = tid + 512;
  const float x0 = xr[i0], x1 = xr[i1], x2 = xr[i2];
  const float m0 = mr[i0], m1 = mr[i1], m2 = mr[i2];
  const float b0 = bias[i0], b1 = bias[i1], b2 = bias[i2];
  float sx2 = x0 * x0 + x1 * x1 + x2 * x2;
  float sm2 = m0 * m0 + m1 * m1 + m2 * m2;
  float smb = m0 * b0 + m1 * b1 + m2 * b2;
  float sb2 = b0 * b0 + b1 * b1 + b2 * b2;
#pragma unroll
  for (int off = 16; off > 0; off >>= 1) {  // wave32 reduction
    sx2 += __shfl_down(sx2, off, 32);
    sm2 += __shfl_down(sm2, off, 32);
    smb += __shfl_down(smb, off, 32);
    sb2 += __shfl_down(sb2, off, 32);
  }
  if (lane == 0) {
    red[0][wave] = sx2; red[1][wave] = sm2;
    red[2][wave] = smb; red[3][wave] = sb2;
  }
  __syncthreads();
  if (tid == 0) {
    float a0 = 0.f, a1 = 0.f, a2 = 0.f, a3 = 0.f;
#pragma unroll
    for (int w = 0; w < 8; ++w) {
      a0 += red[0][w]; a1 += red[1][w]; a2 += red[2][w]; a3 += red[3][w];
    }
    const float x_norm = fminf(fmaxf(sqrtf(a0), 1e-15f), 1.0f - 1e-5f);
    const float mx_norm = fmaxf(sqrtf(a1), 1e-15f);
    const float tt = tanhf(mx_norm / x_norm * atanhf(x_norm));
    const float scale = tt / mx_norm;  // out_i = scale * mx_i ; ||out|| = tt
    const float out2 = tt * tt;
    const float xy = scale * a2;       // <out, b>
    sc[0] = scale;
    sc[1] = 1.0f + 2.0f * xy + a3;                          // coeff of out
    sc[2] = 1.0f - out2;                                    // coeff of b
    sc[3] = 1.0f / (1.0f + 2.0f * xy + out2 * a3 + 1e-15f);
  }
  __syncthreads();
  const float scale = sc[0], co = sc[1], cb = sc[2], idn = sc[3];
  float* orow = Out + (size_t)row * E;
  orow[i0] = (co * (scale * m0) + cb * b0) * idn;
  orow[i1] = (co * (scale * m1) + cb * b1) * idn;
  orow[i2] = (co * (scale * m2) + cb * b2) * idn;
}

// --------------------------------------------------------------------------
// Hyperbolic attention for one (batch, head, 16-query tile).
// grid = (13, 12, 4), block = 128 threads (4 waves).
// --------------------------------------------------------------------------
__global__ __launch_bounds__(128) void attn_kernel(
    const float* __restrict__ Q, const float* __restrict__ K,
    const float* __restrict__ V, const float* __restrict__ hs,
    float* __restrict__ O) {
  __shared__ float Qs[16][KPAD];
  __shared__ float Ks[4][16][KPAD];  // one buffer per wave
  __shared__ float Sc[16][209];      // score strip, odd stride
  __shared__ float q2[16];
  __shared__ float k2s[4][16];
  const int qt = blockIdx.x, h = blockIdx.y, b = blockIdx.z;
  const int tid = threadIdx.x, lane = tid & 31, wave = tid >> 5;
  const size_t base = ((size_t)b * S) * E + (size_t)h * HD;

  // stage Q tile (16 x 64) via async-to-LDS; clamp rows past S (garbage rows
  // are never stored; their q2 feeds only scores of unstored rows).
  for (int i = tid * 4; i < 16 * HD; i += 128 * 4) {  // 2 iterations
    const int m = i >> 6, d = i & 63;
    int s = qt * 16 + m;
    s = (s < S) ? s : (S - 1);
    async_b128(&Q[base + (size_t)s * E + d], &Qs[m][d]);
  }
  wait_async_0();
  __syncthreads();
  if (tid < 16) {
    float a = 0.0f;
    for (int d = 0; d < HD; ++d) { const float t = Qs[tid][d]; a += t * t; }
    q2[tid] = a;
  }
  __syncthreads();

  const float inv_hs = 1.0f / (hs[h] * 8.0f);  // sqrt(HD) = 8
  const int mA = lane & 15;
  const int nB = lane & 15;
  const int koff = (lane >> 4) * 2;

  // ---- scores: each wave owns key tiles j, j+4, ... (private LDS buffers,
  //      only intra-wave ordering needed).
  for (int j = wave; j < NT; j += 4) {
    for (int i = lane * 4; i < 16 * HD; i += 32 * 4) {  // 8 async/wave
      const int n = i >> 6, d = i & 63;
      int s = j * 16 + n;
      s = (s < S) ? s : (S - 1);  // clamp; cols >= S get NEG_BIG below
      async_b128(&K[base + (size_t)s * E + d], &Ks[wave][n][d]);
    }
    wait_async_0();
    if (lane < 16) {
      float a = 0.0f;
      for (int d = 0; d < HD; ++d) {
        const float t = Ks[wave][lane][d];
        a += t * t;
      }
      k2s[wave][lane] = a;
    }
    lds_fence();
    v8f c = {};
#pragma unroll
    for (int kk = 0; kk < HD; kk += 4) {  // 16 WMMAs: dot = q . k
      v2f a, bb;
      a.x = Qs[mA][kk + koff];
      a.y = Qs[mA][kk + koff + 1];
      bb.x = Ks[wave][nB][kk + koff];
      bb.y = Ks[wave][nB][kk + koff + 1];
      c = wmma_f32_4(a, bb, c);
    }
    const int kidx = j * 16 + nB;
    const float k2v = k2s[wave][nB];
#pragma unroll
    for (int r = 0; r < 8; ++r) {
      const int m = ((lane >> 4) * 8) + r;
      float scv;
      if (kidx < S) {
        const float dsq = q2[m] + k2v - 2.0f * c[r];  // ||q-k||^2
        const float den = (1.0f - q2[m]) * (1.0f - k2v) + 1e-15f;
        const float aa = fmaxf(1.0f + 2.0f * dsq / den, 1.0f + 1e-7f);
        const float dist = logf(aa + sqrtf(aa * aa - 1.0f));  // arccosh
        scv = -(dist * dist) * inv_hs;
      } else {
        scv = NEG_BIG;  // padded key -> prob 0
      }
      Sc[m][kidx] = scv;
    }
  }
  __syncthreads();

  // ---- softmax, one row per thread (rows are only 208 wide)
  if (tid < 16) {
    float mx = NEG_BIG;
    for (int kc = 0; kc < NKP; ++kc) mx = fmaxf(mx, Sc[tid][kc]);
    float sum = 0.0f;
    for (int kc = 0; kc < NKP; ++kc) {
      const float e = __expf(Sc[tid][kc] - mx);
      Sc[tid][kc] = e;
      sum += e;
    }
    const float inv = 1.0f / sum;
    for (int kc = 0; kc < NKP; ++kc) Sc[tid][kc] *= inv;
  }
  __syncthreads();

  // ---- out = attn @ V ; wave w owns d-columns [16w, 16w+16)
  const int dcol = wave * 16 + (lane & 15);
  v8f o = {};
  for (int kk = 0; kk < NKP; kk += 4) {  // 52 WMMAs over key dim
    const int kb = kk + koff;
    v2f a, bb;
    a.x = Sc[mA][kb];
    a.y = Sc[mA][kb + 1];
    bb.x = (kb < S) ? V[base + (size_t)kb * E + dcol] : 0.0f;
    bb.y = (kb + 1 < S) ? V[base + (size_t)(kb + 1) * E + dcol] : 0.0f;
    o = wmma_f32_4(a, bb, o);
  }
  const int rb = (lane >> 4) * 8;
#pragma unroll
  for (int r = 0; r < 8; ++r) {
    const int s = qt * 16 + rb + r;
    if (s < S) O[base + (size_t)s * E + dcol] = o[r];
  }
}

// --------------------------------------------------------------------------
extern "C" void kernel_launch(void* const* d_in, const int* in_sizes, int n_in,
                              void* d_out, int out_size, void* d_ws,
                              size_t ws_size, hipStream_t stream) {
  (void)in_sizes; (void)n_in; (void)out_size; (void)ws_size;
  const float* x  = (const float*)d_in[0];
  const float* Wq = (const float*)d_in[1];
  const float* bq = (const float*)d_in[2];
  const float* Wk = (const float*)d_in[3];
  const float* bk = (const float*)d_in[4];
  const float* Wv = (const float*)d_in[5];
  const float* bv = (const float*)d_in[6];
  const float* Wo = (const float*)d_in[7];
  const float* bo = (const float*)d_in[8];
  const float* hs = (const float*)d_in[9];
  float* out = (float*)d_out;
  float* ws = (float*)d_ws;

  const size_t SZ = (size_t)MROWS * E;  // 605,184 floats; ws needs ~9.3 MiB
  float* q  = ws;
  float* k  = ws + SZ;
  float* v  = ws + 2 * SZ;
  float* ao = ws + 3 * SZ;
  float* mo = q;  // q is dead after attention; reuse for Wo GEMM result

  const dim3 gGemm(E / 32, (MROWS + 31) / 32);  // 24 x 25
  gemm_xwT_kernel<<<gGemm, 128, 0, stream>>>(x, Wq, q);
  gemm_xwT_kernel<<<gGemm, 128, 0, stream>>>(x, Wk, k);
  gemm_xwT_kernel<<<gGemm, 128, 0, stream>>>(x, Wv, v);

  mobius_kernel<<<MROWS, 256, 0, stream>>>(x, q, bq, q);
  mobius_kernel<<<MROWS, 256, 0, stream>>>(x, k, bk, k);
  mobius_kernel<<<MROWS, 256, 0, stream>>>(x, v, bv, v);

  attn_kernel<<<dim3(NT, H, B), 128, 0, stream>>>(q, k, v, hs, ao);

  gemm_xwT_kernel<<<gGemm, 128, 0, stream>>>(ao, Wo, mo);
  mobius_kernel<<<MROWS, 256, 0, stream>>>(ao, mo, bo, out);
}